// FIS_ImportanceAssessment_463856468232
// MI455X (gfx1250) — compile-verified
//
#include <hip/hip_runtime.h>
#include <hip/hip_bf16.h>

typedef float v2f __attribute__((ext_vector_type(2)));
typedef float v8f __attribute__((ext_vector_type(8)));

#define C_CH 256
#define HW_SZ 4096              // 64*64
#define PIX_PER_BLOCK 64
#define THREADS 256
#define CH_SPLIT 4
#define CH_PER_THREAD (C_CH / CH_SPLIT)   // 64

__global__ __launch_bounds__(THREADS)
void fis_importance_kernel(const float* __restrict__ features,
                           const float* __restrict__ W1,
                           const float* __restrict__ b1,
                           const float* __restrict__ W2,
                           const float* __restrict__ b2,
                           float* __restrict__ out)
{
    __shared__ float s_sum[THREADS];
    __shared__ float s_sq[THREADS];
    __shared__ float s_h[PIX_PER_BLOCK * 17];   // padded pitch 17: no bank conflicts

    const int tid = threadIdx.x;
    const int p   = tid & (PIX_PER_BLOCK - 1);  // pixel within block
    const int g   = tid >> 6;                   // channel group 0..3
    const int pixelBase = blockIdx.x * PIX_PER_BLOCK;
    const int pixel = pixelBase + p;            // flat (b,h,w)
    const int b  = pixel >> 12;                 // / 4096
    const int hw = pixel & (HW_SZ - 1);

    // ---- channel reduction: 4 threads per pixel, 64 channels each ----
    const float* fp = features + ((size_t)(b * C_CH + g * CH_PER_THREAD)) * HW_SZ + hw;
    float sum = 0.0f, sq = 0.0f;
#pragma unroll 16
    for (int c = 0; c < CH_PER_THREAD; ++c) {
        float v = fp[(size_t)c * HW_SZ];        // coalesced 128B per wave, 16KB stride
        sum += v;
        sq = fmaf(v, v, sq);
    }
    s_sum[tid] = sum;
    s_sq[tid]  = sq;
    __syncthreads();

    if (tid < PIX_PER_BLOCK) {                  // waves 0,1 fully active -> EXEC all ones
        float fsum = s_sum[p] + s_sum[p + 64] + s_sum[p + 128] + s_sum[p + 192];
        float fsq  = s_sq[p]  + s_sq[p + 64]  + s_sq[p + 128]  + s_sq[p + 192];

        const float invC = 1.0f / (float)C_CH;
        float sq_mean = fsq * invC;
        float mag = sqrtf(sq_mean);
        float var = (fsq - fsum * fsum * invC) * (1.0f / (float)(C_CH - 1)); // ddof=1
        var = fmaxf(var, 0.0f);
        float stdv = sqrtf(var);
        float t_mag  = fminf(fmaxf(mag, 0.0f), 1.0f);
        float t_var  = fminf(var,  1.0f);
        float t_grad = fminf(stdv, 1.0f);

        const int lane = tid & 31;
        const int half = lane >> 4;             // 0: lanes 0-15 (K=0,1), 1: lanes 16-31 (K=2,3)
        const int l15  = lane & 15;
        const int waveBase = tid & 32;          // wave-local pixel base within block (0 or 32)

        // B (4x16): rows = [W1 row0; W1 row1; W1 row2; b1]; vgpr0={row0|row2}, vgpr1={row1|row3}
        v2f bm;
        bm.x = half ? W1[2 * 16 + l15] : W1[0 * 16 + l15];
        bm.y = half ? b1[l15]          : W1[1 * 16 + l15];

#pragma unroll
        for (int s = 0; s < 2; ++s) {           // two 16-pixel subtiles per wave
            int src = s * 16 + l15;             // wave-local pixel owning row M=l15
            float m  = __shfl(t_mag,  src, 32);
            float v  = __shfl(t_var,  src, 32);
            float gr = __shfl(t_grad, src, 32);
            // A (16x4): lanes 0-15 hold [K0,K1]=[mag,var]; lanes 16-31 hold [K2,K3]=[std,1]
            v2f a;
            a.x = half ? gr   : m;
            a.y = half ? 1.0f : v;
            v8f cz = {};
            v8f d = __builtin_amdgcn_wmma_f32_16x16x4_f32(
                false, a, false, bm, (short)0, cz, false, false);
            // ReLU + transpose D (pixels x features) into LDS
            int mrow = waveBase + s * 16 + (half ? 8 : 0);
#pragma unroll
            for (int r = 0; r < 8; ++r)
                s_h[(mrow + r) * 17 + l15] = fmaxf(d[r], 0.0f);
        }

        // ---- layer 2: 16->1 dot + sigmoid (within-wave DS ops are in-order) ----
        float acc = b2[0];
#pragma unroll
        for (int n = 0; n < 16; ++n)
            acc = fmaf(s_h[tid * 17 + n], W2[n], acc);
        out[pixel] = 1.0f / (1.0f + __expf(-acc));
    }
}

extern "C" void kernel_launch(void* const* d_in, const int* in_sizes, int n_in,
                              void* d_out, int out_size, void* d_ws, size_t ws_size,
                              hipStream_t stream) {
    const float* features = (const float*)d_in[0];
    const float* W1 = (const float*)d_in[1];
    const float* b1 = (const float*)d_in[2];
    const float* W2 = (const float*)d_in[3];
    const float* b2 = (const float*)d_in[4];
    float* out = (float*)d_out;

    const int total_pixels = 16 * 64 * 64;                 // 65536
    const int blocks = total_pixels / PIX_PER_BLOCK;       // 1024
    fis_importance_kernel<<<blocks, THREADS, 0, stream>>>(features, W1, b1, W2, b2, out);
}